// heatmap_loss5_26113401160133
// MI455X (gfx1250) — compile-verified
//
#include <hip/hip_runtime.h>
#include <hip/hip_bf16.h>

// MI455X / gfx1250, wave32. Streaming bandwidth-bound loss kernel.
// Pass 1: one 256-thread block per sample, float4-coalesced read of Yp/Yg,
//         building row/col "any>0.5" bitmasks + sums. Block sums reduced via
//         v_wmma_f32_16x16x4_f32 with an all-ones B matrix (layout-proof sum).
// Pass 2 (gen-mask branch only): bounded superellipse rectangle re-read.
// Finalize: tiny kernel reduces 512 ratios (again via WMMA) -> log(total)/B.

static constexpr int LDIM = 256;
static constexpr int LL2  = LDIM * LDIM;

typedef float v2f __attribute__((ext_vector_type(2)));
typedef float v8f __attribute__((ext_vector_type(8)));

// Exact fp32 block-sum of 256 values using V_WMMA_F32_16X16X4_F32.
// B = all-ones => D row-sums accumulate every element of A regardless of the
// A-matrix lane/VGPR mapping, so the total is the exact sum of all 256 inputs.
__device__ __forceinline__ float block_reduce_wmma(float v, float* buf) {
    __syncthreads();                 // protect buf reuse across calls
    buf[threadIdx.x] = v;
    __syncthreads();
    if (threadIdx.x < 32) {          // uniform branch: wave 0 fully active (EXEC all-1s)
        const int m  = threadIdx.x & 15;
        const int kh = (threadIdx.x >> 4) & 1;
        v8f acc = {};
        v2f ones; ones[0] = 1.0f; ones[1] = 1.0f;
#pragma unroll
        for (int b = 0; b < 4; ++b) {
            v2f a;
            a[0] = buf[b * 64 + m * 4 + 2 * kh + 0];
            a[1] = buf[b * 64 + m * 4 + 2 * kh + 1];
            acc = __builtin_amdgcn_wmma_f32_16x16x4_f32(
                false, a, false, ones, (short)0, acc, false, false);
        }
        // lane<16 holds rows 0-7 (col = lane), lane>=16 holds rows 8-15.
        float s = ((acc[0] + acc[1]) + (acc[2] + acc[3])) +
                  ((acc[4] + acc[5]) + (acc[6] + acc[7]));
        s += __shfl_xor(s, 16, 32);
        if (threadIdx.x == 0) buf[0] = s;
    }
    __syncthreads();
    return buf[0];
}

__global__ __launch_bounds__(256) void hm_loss_sample_kernel(
    const float* __restrict__ Ypred, const float* __restrict__ Ygt,
    float* __restrict__ ratios)
{
    __shared__ unsigned rowMask[8];   // 256 row bits: any(Yp[r,*] > 0.5)
    __shared__ unsigned colMask[8];   // 256 col bits: any(Yp[*,c] > 0.5)
    __shared__ int      anyG;         // max(Yg) > 0  (Yg is 0/1)
    __shared__ float    redBuf[256];
    __shared__ float    fpar[2];      // gt0, gt1
    __shared__ int      ipar[6];      // x_r, y_r, rmin, rmax, cmin, cmax

    const int tid = threadIdx.x;
    const int s   = blockIdx.x;
    if (tid < 8) { rowMask[tid] = 0u; colMask[tid] = 0u; }
    if (tid == 0) anyG = 0;
    __syncthreads();

    const float4* Yp4 = (const float4*)(Ypred + (size_t)s * LL2);
    const float4* Yg4 = (const float4*)(Ygt   + (size_t)s * LL2);

    float sumP = 0.f, sumPG = 0.f;
    bool  c0 = false, c1 = false, c2 = false, c3 = false;  // per-thread fixed 4 columns
    bool  gAny = false;
    const int lane = tid & 31;

#pragma unroll 4
    for (int i = 0; i < 64; ++i) {
        const int idx4 = i * 256 + tid;       // block reads 4 full rows / iter, coalesced b128
        const float4 p = Yp4[idx4];
        const float4 g = Yg4[idx4];
        sumP  += (p.x + p.y) + (p.z + p.w);
        sumPG += (g.x * p.x + g.y * p.y) + (g.z * p.z + g.w * p.w);
        gAny = gAny | (g.x > 0.f) | (g.y > 0.f) | (g.z > 0.f) | (g.w > 0.f);
        const bool b0 = p.x > 0.5f, b1 = p.y > 0.5f, b2 = p.z > 0.5f, b3 = p.w > 0.5f;
        c0 |= b0; c1 |= b1; c2 |= b2; c3 |= b3;
        // every wave's 32 lanes lie in one row here (32*4=128 consecutive elems)
        const unsigned bal = __builtin_amdgcn_ballot_w32(b0 | b1 | b2 | b3);
        if (bal && lane == 0) {
            const int r = idx4 >> 6;
            atomicOr(&rowMask[r >> 5], 1u << (r & 31));
        }
    }
    {   // column bits: thread tid always touches columns cbase..cbase+3
        const int cbase = (tid & 63) * 4;
        const unsigned bits =
            ((unsigned)c0 | ((unsigned)c1 << 1) | ((unsigned)c2 << 2) | ((unsigned)c3 << 3))
            << (cbase & 31);
        if (bits) atomicOr(&colMask[cbase >> 5], bits);
    }
    if (gAny) atomicOr(&anyG, 1);

    const float SP  = block_reduce_wmma(sumP,  redBuf);   // sum(Yp)
    const float SPG = block_reduce_wmma(sumPG, redBuf);   // sum(Yg*Yp)

    if (tid == 0) {
        int rFirst = -1, rLast = -1, cFirst = -1, cLast = -1;
        for (int w = 0; w < 8; ++w) {
            unsigned m = rowMask[w];
            if (m) { if (rFirst < 0) rFirst = w * 32 + __builtin_ctz(m);
                     rLast = w * 32 + 31 - __builtin_clz(m); }
            m = colMask[w];
            if (m) { if (cFirst < 0) cFirst = w * 32 + __builtin_ctz(m);
                     cLast = w * 32 + 31 - __builtin_clz(m); }
        }
        // argmax(all-false) == 0 semantics
        const int left  = (rFirst < 0) ? 0 : rFirst;
        const int right = (rFirst < 0) ? 0 : (LDIM - 1 - rLast);
        const int up    = (cFirst < 0) ? 0 : cFirst;
        const int down  = (cFirst < 0) ? 0 : (LDIM - 1 - cLast);
        const int x_r = (right - left) >> 1;     // floor division (matches //)
        const int y_r = (down  - up)   >> 1;
        const float gt0 = (float)(left + x_r);   // compared vs COLUMN coord (ref's cross-wiring)
        const float gt1 = (float)(up + y_r);     // compared vs ROW coord
        int rmin = 1, rmax = 0, cmin = 1, cmax = 0;   // empty default
        if (!anyG && x_r != 0 && y_r != 0) {
            // mask>0.1 requires d0^4+d1^4 < ln(10)*4/9 = 1.02337 => |d| < 1.00580
            const float hwC = 1.0062f * fabsf((float)x_r);
            const float hwR = 1.0062f * fabsf((float)y_r);
            cmin = max(0,        (int)ceilf (gt0 - hwC) - 1);
            cmax = min(LDIM - 1, (int)floorf(gt0 + hwC) + 1);
            rmin = max(0,        (int)ceilf (gt1 - hwR) - 1);
            rmax = min(LDIM - 1, (int)floorf(gt1 + hwR) + 1);
        }
        fpar[0] = gt0; fpar[1] = gt1;
        ipar[0] = x_r; ipar[1] = y_r;
        ipar[2] = rmin; ipar[3] = rmax; ipar[4] = cmin; ipar[5] = cmax;
    }
    __syncthreads();

    float posGen = 0.f;
    {
        const int rmin = ipar[2], rmax = ipar[3], cmin = ipar[4], cmax = ipar[5];
        if (rmax >= rmin && cmax >= cmin) {   // uniform condition
            const float gt0 = fpar[0], gt1 = fpar[1];
            const float fxr = (float)ipar[0], fyr = (float)ipar[1];
            const int   ncols = cmax - cmin + 1;
            const int   npix  = (rmax - rmin + 1) * ncols;
            const float* Yp = Ypred + (size_t)s * LL2;
            for (int k = tid; k < npix; k += 256) {
                const int q = k / ncols;
                const int r = rmin + q;
                const int c = cmin + (k - q * ncols);
                const float p  = Yp[r * LDIM + c];
                const float d0 = ((float)c - gt0) / fxr;
                const float d1 = ((float)r - gt1) / fyr;
                const float d0s = d0 * d0, d1s = d1 * d1;
                const float h = d0s * d0s + d1s * d1s;      // d0^4 + d1^4
                if (expf(-2.25f * h) > 0.1f) posGen += p;   // exp(-h/(2*sigma)^2) > MASK_T
            }
        }
    }
    const float PGEN = block_reduce_wmma(posGen, redBuf);

    if (tid == 0) {
        const float positive = anyG ? SPG : PGEN;
        const float negative = SP - positive;               // sum((1-M)*Yp), M in {0,1}
        ratios[s] = negative / (positive + 1e-6f);
    }
}

__global__ __launch_bounds__(256) void hm_loss_final_kernel(
    const float* __restrict__ ratios, float* __restrict__ out, int B)
{
    __shared__ float redBuf[256];
    float v = 0.f;
    for (int i = threadIdx.x; i < B; i += 256) v += ratios[i];
    const float total = block_reduce_wmma(v, redBuf);
    if (threadIdx.x == 0)
        out[0] = (total == 0.0f) ? 0.0f : (logf(total) / (float)B);
}

extern "C" void kernel_launch(void* const* d_in, const int* in_sizes, int n_in,
                              void* d_out, int out_size, void* d_ws, size_t ws_size,
                              hipStream_t stream) {
    const float* Yp = (const float*)d_in[0];
    const float* Yg = (const float*)d_in[1];
    float* out    = (float*)d_out;
    float* ratios = (float*)d_ws;                 // B floats of scratch
    const int B = in_sizes[0] / LL2;
    hm_loss_sample_kernel<<<B, 256, 0, stream>>>(Yp, Yg, ratios);
    hm_loss_final_kernel<<<1, 256, 0, stream>>>(ratios, out, B);
}